// GeneralizedCrossEntropy_32959579029758
// MI455X (gfx1250) — compile-verified
//
#include <hip/hip_runtime.h>
#include <hip/hip_bf16.h>
#include <math.h>

// GCE: loss_i = (1 - p_t^q)/q,  p_t = exp(logit_t - logsumexp(row)),  q = 0.8
#define GCE_Q    0.8f
#define GCE_INVQ 1.25f

constexpr int BLOCK      = 256;    // 8 waves (wave32)
constexpr int TILE_ELEMS = 4096;   // 16 KB fp32 tile staged via TDM

#if defined(__HIP_DEVICE_COMPILE__) && __has_builtin(__builtin_amdgcn_tensor_load_to_lds)
#define GCE_TDM 1
#else
#define GCE_TDM 0
#endif

__device__ __forceinline__ void gce_merge(float& m, float& s, float om, float os) {
    float nm = fmaxf(m, om);
    s = s * __expf(m - nm) + os * __expf(om - nm);
    m = nm;
}

#if GCE_TDM
typedef unsigned int gce_u32x4 __attribute__((ext_vector_type(4)));
typedef int          gce_i32x4 __attribute__((ext_vector_type(4)));
typedef int          gce_i32x8 __attribute__((ext_vector_type(8)));

// 1-D tensor-DMA: copy `elems` fp32 from global to LDS (elems <= 65535).
// D# layout per CDNA5 ISA ch.8: group0 = {count/type, lds_addr, global_addr},
// group1 = {data_size, tensor_dim0, tile_dim0, tensor_dim0_stride}.
__device__ __forceinline__ void gce_tdm_load(const float* gsrc, void* ldst, int elems) {
    unsigned long long ga = (unsigned long long)(uintptr_t)gsrc;
    unsigned int       la = (unsigned int)(uintptr_t)ldst;  // low 32b of flat addr == LDS offset
    gce_u32x4 g0;
    g0[0] = 1u;                                           // count=1, user descriptor
    g0[1] = la;                                           // lds_addr
    g0[2] = (unsigned int)(ga & 0xFFFFFFFFull);           // global_addr[31:0]
    g0[3] = ((unsigned int)(ga >> 32) & 0x01FFFFFFu)      // global_addr[56:32]
            | 0x80000000u;                                // type = 2 ("image")
    unsigned int e = (unsigned int)elems;
    gce_i32x8 g1;
    g1[0] = 0x00020000;                 // workgroup_mask=0, data_size=2 (4 bytes)
    g1[1] = (int)((e & 0xFFFFu) << 16); // tensor_dim0[15:0]  @ bits 63:48
    g1[2] = (int)((e >> 16) & 0xFFFFu); // tensor_dim0[31:16] @ bits 79:64 ; tensor_dim1 = 0
    g1[3] = (int)((e & 0xFFFFu) << 16); // tile_dim0          @ bits 127:112
    g1[4] = 0;                          // tile_dim1 = tile_dim2 = 0 (unused)
    g1[5] = (int)e;                     // tensor_dim0_stride[31:0]
    g1[6] = 0;
    g1[7] = 0;
    gce_i32x4 z4 = {0, 0, 0, 0};              // groups 2/3 unused (<=2D tensor)
    gce_i32x8 z8 = {0, 0, 0, 0, 0, 0, 0, 0};
    // 6-arg form (this toolchain): (g0 v4u32, g1 v8i32, v4i32, v4i32, v8i32, cpol)
    __builtin_amdgcn_tensor_load_to_lds(g0, g1, z4, z4, z8, 0);
}

__device__ __forceinline__ void gce_wait_tensorcnt_le(int n) {
#if __has_builtin(__builtin_amdgcn_s_wait_tensorcnt)
    if (n == 0) __builtin_amdgcn_s_wait_tensorcnt(0);
    else        __builtin_amdgcn_s_wait_tensorcnt(1);
#else
    if (n == 0) asm volatile("s_wait_tensorcnt 0x0" ::: "memory");
    else        asm volatile("s_wait_tensorcnt 0x1" ::: "memory");
#endif
}
#endif  // GCE_TDM

__global__ void GeneralizedCrossEntropy_32959579029758_kernel(
    const float* __restrict__ logits,
    const long long* __restrict__ targets,
    float* __restrict__ out, int C) {
    const int row  = blockIdx.x;
    const int tid  = threadIdx.x;
    const int wave = tid >> 5;
    const int lane = tid & 31;
    const float* rowp = logits + (size_t)row * (size_t)C;

    __shared__ float red_m[BLOCK / 32];
    __shared__ float red_s[BLOCK / 32];

    float m = -INFINITY;
    float s = 0.0f;

#if GCE_TDM
    __shared__ __align__(16) float tile[2][TILE_ELEMS];
    const int ntiles = (C + TILE_ELEMS - 1) / TILE_ELEMS;

    // TDM ignores EXEC -> make the leader test wave-uniform so the compiler
    // emits a scalar branch around the issue/wait block.
    int leader = (tid < 32) ? 1 : 0;
    leader = __builtin_amdgcn_readfirstlane(leader);

    if (leader) {
        int e0 = (C < TILE_ELEMS) ? C : TILE_ELEMS;
        gce_tdm_load(rowp, &tile[0][0], e0);
    }

    for (int t = 0; t < ntiles; ++t) {
        const int buf   = t & 1;
        const int base  = t * TILE_ELEMS;
        const int elems = (C - base < TILE_ELEMS) ? (C - base) : TILE_ELEMS;
        if (leader) {
            if (t + 1 < ntiles) {
                const int nbase  = base + TILE_ELEMS;
                const int nelems = (C - nbase < TILE_ELEMS) ? (C - nbase) : TILE_ELEMS;
                gce_tdm_load(rowp + nbase, &tile[buf ^ 1][0], nelems);
                gce_wait_tensorcnt_le(1);   // oldest (tile t) has landed
            } else {
                gce_wait_tensorcnt_le(0);
            }
        }
        __syncthreads();                    // tile t visible to all waves

        const float4* lb = (const float4*)(&tile[buf][0]);
        const int nvec = elems >> 2;
        for (int v = tid; v < nvec; v += BLOCK) {
            float4 x = lb[v];
            float mx = fmaxf(fmaxf(x.x, x.y), fmaxf(x.z, x.w));
            float nm = fmaxf(m, mx);
            s = s * __expf(m - nm)
              + __expf(x.x - nm) + __expf(x.y - nm)
              + __expf(x.z - nm) + __expf(x.w - nm);
            m = nm;
        }
        if (tid == 0) {                     // generic tail (elems % 4)
            for (int e2 = nvec << 2; e2 < elems; ++e2) {
                float x = tile[buf][e2];
                float nm = fmaxf(m, x);
                s = s * __expf(m - nm) + __expf(x - nm);
                m = nm;
            }
        }
        __syncthreads();                    // compute(t) done before buf^1 reuse
    }
#else
    // Fallback: stream directly from global with B128 loads.
    const float4* rp4 = (const float4*)rowp;
    const int nvec = C >> 2;
    for (int v = tid; v < nvec; v += BLOCK) {
        float4 x = rp4[v];
        float mx = fmaxf(fmaxf(x.x, x.y), fmaxf(x.z, x.w));
        float nm = fmaxf(m, mx);
        s = s * __expf(m - nm)
          + __expf(x.x - nm) + __expf(x.y - nm)
          + __expf(x.z - nm) + __expf(x.w - nm);
        m = nm;
    }
    if (tid == 0) {
        for (int e2 = nvec << 2; e2 < C; ++e2) {
            float x = rowp[e2];
            float nm = fmaxf(m, x);
            s = s * __expf(m - nm) + __expf(x - nm);
            m = nm;
        }
    }
#endif

    // Wave-level (m,s) reduction, wave32.
    for (int off = 16; off > 0; off >>= 1) {
        float om = __shfl_xor(m, off, 32);
        float os = __shfl_xor(s, off, 32);
        gce_merge(m, s, om, os);
    }
    if (lane == 0) { red_m[wave] = m; red_s[wave] = s; }
    __syncthreads();

    if (tid == 0) {
        float M = red_m[0], S = red_s[0];
        #pragma unroll
        for (int w = 1; w < BLOCK / 32; ++w) gce_merge(M, S, red_m[w], red_s[w]);
        long long tgt = targets[row];
        float logit_t = rowp[tgt];
        float log_p   = logit_t - (M + __logf(S));       // log p_t
        out[row] = (1.0f - __expf(GCE_Q * log_p)) * GCE_INVQ;
    }
}

extern "C" void kernel_launch(void* const* d_in, const int* in_sizes, int n_in,
                              void* d_out, int out_size, void* d_ws, size_t ws_size,
                              hipStream_t stream) {
    const float*     logits  = (const float*)d_in[0];
    const long long* targets = (const long long*)d_in[1];
    float*           out     = (float*)d_out;
    const int N = in_sizes[1];              // 8192 rows
    const int C = in_sizes[0] / N;          // 32000 classes
    hipLaunchKernelGGL(GeneralizedCrossEntropy_32959579029758_kernel,
                       dim3(N), dim3(BLOCK), 0, stream,
                       logits, targets, out, C);
}